// loss_generator_73529840107705
// MI455X (gfx1250) — compile-verified
//
#include <hip/hip_runtime.h>

// ---------------------------------------------------------------------------
// Gray-Scott reaction-diffusion residual (f_u, f_v) for (48,1,96,96,96) fp32.
// Memory-bound 13-point stencil. CDNA5 path: TDM tensor_load_to_lds async DMA
// of halo tiles into LDS (TENSORcnt), LDS stencil, non-temporal output stores.
// ---------------------------------------------------------------------------

#define NTIME 50
#define NY 100
#define NX 100
#define NZ 100
#define OUT_T 48
#define OUT_D 96

#define TY 4      // outputs per block in y
#define TX 4      // outputs per block in x
#define HY 8      // halo tile y extent (TY + 4)
#define HX 8      // halo tile x extent (TX + 4)
#define HZ 100    // halo tile z extent (full line: 96 + 4)
#define TILE_ELEMS (HY * HX * HZ)   // 6400 floats = 25.6 KB per channel

typedef unsigned int u32x4 __attribute__((ext_vector_type(4)));
typedef int          i32x8 __attribute__((ext_vector_type(8)));
typedef int          i32x4 __attribute__((ext_vector_type(4)));

#if defined(__gfx1250__) && __has_builtin(__builtin_amdgcn_tensor_load_to_lds)
#define HAVE_TDM 1
#else
#define HAVE_TDM 0
#endif

#if HAVE_TDM
// Issue one TDM DMA: 3-D tile (z=100 contiguous, x=8 @ stride 100, y=8 @ stride
// 10000) from global to LDS at byte offset lds_byte. D# layout per CDNA5 ISA
// ch. 8 (groups 0-3). No OOB: tile lies strictly inside the 100^3 volume.
// This toolchain's builtin takes 6 args: (g0 v4u, g1 v8i, g2 v4i, g3 v4i,
// extra v8i, cpol i32) -- extra operand passed zero-filled (probe-verified
// pattern).
__device__ __forceinline__ void tdm_load_tile_3d(const float* gsrc, unsigned lds_byte) {
  unsigned long long ga = (unsigned long long)gsrc;
  u32x4 g0;
  g0[0] = 1u;                                              // count=1, user mode, no gather
  g0[1] = lds_byte;                                        // lds_addr (bytes)
  g0[2] = (unsigned)ga;                                    // global_addr[31:0]
  g0[3] = ((unsigned)(ga >> 32) & 0x01FFFFFFu) | (2u << 30); // global_addr[56:32], type=2
  i32x8 g1;
  g1[0] = 0x00020000;                  // workgroup_mask=0, data_size=2 (4 bytes)
  g1[1] = (int)(NZ << 16);             // tensor_dim0[15:0]=100 (atomic_barrier_addr=0)
  g1[2] = (int)(NX << 16);             // tensor_dim0 hi=0, tensor_dim1[15:0]=100
  g1[3] = (int)(HZ << 16);             // tensor_dim1 hi=0, tile_dim0=100
  g1[4] = (int)(HX | (HY << 16));      // tile_dim1=8, tile_dim2=8
  g1[5] = NZ;                          // tensor_dim0_stride lo32 = 100
  g1[6] = (int)((NZ * NX) << 16);      // stride0 hi=0, tensor_dim1_stride lo16 = 10000
  g1[7] = (int)((unsigned)(NZ * NX) >> 16);  // tensor_dim1_stride[47:16] = 0
  i32x4 g2;
  g2[0] = NY;                          // tensor_dim2 = 100
  g2[1] = 0;                           // tensor_dim3 (unused)
  g2[2] = NZ * NX * NY;                // tensor_dim2_stride lo32 (unused for 3D tile)
  g2[3] = 0;                           // stride hi, tile_dim3=0 (unused)
  i32x4 g3 = (i32x4){0, 0, 0, 0};      // dims 3/4 unused
  i32x8 gx = (i32x8){0, 0, 0, 0, 0, 0, 0, 0};  // uncharacterized extra operand
  __builtin_amdgcn_tensor_load_to_lds(g0, g1, g2, g3, gx, 0);
}
#endif

__global__ __launch_bounds__(192)
void gs_residual_kernel(const float* __restrict__ in, float* __restrict__ out) {
  __shared__ float tu[TILE_ELEMS];
  __shared__ float tv[TILE_ELEMS];

  const int t  = blockIdx.z;          // 0..47
  const int y0 = blockIdx.y * TY;     // interior tile origin (0..92)
  const int x0 = blockIdx.x * TX;

  const size_t cstride = (size_t)NY * NX * NZ;  // 1,000,000 elems per channel-slice
  // Halo tile origin in full coords is (y0, x0, 0): interior +2 and halo -2 cancel.
  const float* srcU = in + ((size_t)t * 2 + 0) * cstride
                         + (size_t)y0 * (NX * NZ) + (size_t)x0 * NZ;
  const float* srcV = srcU + cstride;

  const int tid = threadIdx.x + threadIdx.y * 96;

#if HAVE_TDM
  // Wave 0 issues both DMAs and waits its own TENSORcnt; others wait at barrier.
  if (__builtin_amdgcn_readfirstlane(tid) == 0) {
    tdm_load_tile_3d(srcU, (unsigned)(unsigned long long)tu);
    tdm_load_tile_3d(srcV, (unsigned)(unsigned long long)tv);
    __builtin_amdgcn_s_wait_tensorcnt(0);
  }
  __syncthreads();
#else
  for (int i = tid; i < TILE_ELEMS; i += 192) {
    int hy = i / (HX * HZ);
    int r  = i - hy * (HX * HZ);
    int hx = r / HZ;
    int hz = r - hx * HZ;
    size_t g = (size_t)hy * (NX * NZ) + (size_t)hx * NZ + hz;
    tu[i] = srcU[g];
    tv[i] = srcV[g];
  }
  __syncthreads();
#endif

  // Stencil weights: per-axis [-1/12, 4/3, -5/2, 4/3, -1/12]; center summed 3x.
  const float s2w = -1.0f / 12.0f;
  const float s1w = 4.0f / 3.0f;
  const float cw  = -7.5f;
  const float inv_dx2 = 0.2304f;   // (48/100)^2 = 1/DX^2
  const float inv_dt  = 2.0f;      // 1/DT

  const int z = threadIdx.x;       // 0..95, contiguous axis
  const float* nextU = in + ((size_t)(t + 1) * 2 + 0) * cstride
                          + (size_t)(y0 + 2) * (NX * NZ) + (size_t)(x0 + 2) * NZ + (z + 2);
  const float* nextV = nextU + cstride;

  const size_t outCh = (size_t)OUT_T * OUT_D * OUT_D * OUT_D;
  float* outU = out;
  float* outV = out + outCh;

#pragma unroll
  for (int j = 0; j < 8; ++j) {
    const int col = threadIdx.y * 8 + j;   // 0..15
    const int ly  = col >> 2;              // 0..3
    const int lx  = col & 3;               // 0..3
    const int ci  = ((ly + 2) * HX + (lx + 2)) * HZ + (z + 2);

    const float uc = tu[ci];
    const float vc = tv[ci];

    float lap_u = s2w * (tu[ci - 2] + tu[ci + 2] +
                         tu[ci - 2 * HZ] + tu[ci + 2 * HZ] +
                         tu[ci - 2 * HX * HZ] + tu[ci + 2 * HX * HZ])
                + s1w * (tu[ci - 1] + tu[ci + 1] +
                         tu[ci - HZ] + tu[ci + HZ] +
                         tu[ci - HX * HZ] + tu[ci + HX * HZ])
                + cw * uc;
    float lap_v = s2w * (tv[ci - 2] + tv[ci + 2] +
                         tv[ci - 2 * HZ] + tv[ci + 2 * HZ] +
                         tv[ci - 2 * HX * HZ] + tv[ci + 2 * HX * HZ])
                + s1w * (tv[ci - 1] + tv[ci + 1] +
                         tv[ci - HZ] + tv[ci + HZ] +
                         tv[ci - HX * HZ] + tv[ci + HX * HZ])
                + cw * vc;
    lap_u *= inv_dx2;
    lap_v *= inv_dx2;

    const size_t noff = (size_t)ly * (NX * NZ) + (size_t)lx * NZ;
    const float un = nextU[noff];
    const float vn = nextV[noff];
    const float u_t = (un - uc) * inv_dt;
    const float v_t = (vn - vc) * inv_dt;
    const float uv2 = uc * vc * vc;

    const float fu = 0.2f * lap_u - uv2 + 0.025f * (1.0f - uc) - u_t;
    const float fv = 0.1f * lap_v + uv2 - 0.08f * vc - v_t;

    const size_t o = ((((size_t)t * OUT_D) + (y0 + ly)) * OUT_D + (x0 + lx)) * OUT_D + z;
    __builtin_nontemporal_store(fu, outU + o);
    __builtin_nontemporal_store(fv, outV + o);
  }
}

extern "C" void kernel_launch(void* const* d_in, const int* in_sizes, int n_in,
                              void* d_out, int out_size, void* d_ws, size_t ws_size,
                              hipStream_t stream) {
  (void)in_sizes; (void)n_in; (void)out_size; (void)d_ws; (void)ws_size;
  const float* in = (const float*)d_in[0];   // (50, 2, 100, 100, 100) fp32
  float* out = (float*)d_out;                // f_u ++ f_v, each (48,1,96,96,96)

  dim3 grid(OUT_D / TX, OUT_D / TY, OUT_T);  // (24, 24, 48)
  dim3 block(96, 2, 1);                      // 6 wave32s
  gs_residual_kernel<<<grid, block, 0, stream>>>(in, out);
}